// ACoef_15633680958283
// MI455X (gfx1250) — compile-verified
//
#include <hip/hip_runtime.h>

// f32 WMMA fragment types (CDNA5 / gfx1250, wave32)
typedef __attribute__((ext_vector_type(2))) float v2f;
typedef __attribute__((ext_vector_type(8))) float v8f;

constexpr int G      = 96;        // matrix dim
constexpr int LDST   = 108;       // padded LDS row stride (dwords):
                                  //  - 108 mod 64 = 44, gcd=4 -> A-frag column reads hit 16 distinct banks
                                  //  - 2*108 mod 64 = 24 -> B-frag half-wave banks disjoint
                                  //  - 8*108 mod 64 = 32 -> C-store half-wave banks disjoint
                                  //  - multiple of 4 -> rows 16B-aligned for b128 ingest
constexpr int NTILE  = G / 16;    // 6 tiles per dim
constexpr int TILES  = NTILE * NTILE;  // 36
constexpr int BLOCK  = 256;       // 8 waves of 32
constexpr int WAVES  = BLOCK / 32;
constexpr int ROWS   = 5;
constexpr int COLS   = 4;
constexpr int MATF   = G * LDST;  // floats per LDS matrix

// C = A * B for 96x96 f32 matrices resident in LDS, via V_WMMA_F32_16X16X4_F32.
// Each wave owns tiles {wave, wave+8, ...} of the 6x6 tile grid.
__device__ __forceinline__ void wmma_gemm96(const float* __restrict__ Am,
                                            const float* __restrict__ Bm,
                                            float* __restrict__ Cm,
                                            int wave, int half, int l) {
  for (int t = wave; t < TILES; t += WAVES) {
    const int mt = (t / NTILE) * 16;
    const int nt = (t % NTILE) * 16;
    v8f acc = {};
    for (int kk = 0; kk < G; kk += 4) {
      const int k0 = kk + 2 * half;   // lanes 0-15 -> K={kk,kk+1}; 16-31 -> K={kk+2,kk+3}
      v2f a, b;
      // A-matrix 16x4: lane holds row M = mt+l, two consecutive K values
      a.x = Am[(mt + l) * LDST + k0];
      a.y = Am[(mt + l) * LDST + k0 + 1];
      // B-matrix 4x16: lane holds column N = nt+l, two consecutive K values
      b.x = Bm[(k0)     * LDST + nt + l];
      b.y = Bm[(k0 + 1) * LDST + nt + l];
      acc = __builtin_amdgcn_wmma_f32_16x16x4_f32(
          /*neg_a=*/false, a, /*neg_b=*/false, b,
          /*c_mod=*/(short)0, acc, /*reuse_a=*/false, /*reuse_b=*/false);
    }
    // C/D layout: VGPR g -> row (g + 8*half), col = lane%16
#pragma unroll
    for (int g = 0; g < 8; ++g) {
      Cm[(mt + g + 8 * half) * LDST + nt + l] = acc[g];
    }
  }
}

__global__ __launch_bounds__(BLOCK) void acoef_kernel(
    const float* __restrict__ x,     // [B, 96, 96]
    const float* __restrict__ coef,  // [5, 4]
    float* __restrict__ out) {       // [B]
  extern __shared__ float lds[];
  float* X  = lds;                 // x
  float* A2 = lds + MATF;          // x^2
  float* A3 = lds + 2 * MATF;      // x^3
  float* T  = lds + 3 * MATF;      // 5 trace accumulators (padded to 8)

  const int tid  = threadIdx.x;
  const int wave = tid >> 5;
  const int lane = tid & 31;
  const int half = lane >> 4;
  const int l    = lane & 15;

  if (tid < 8) T[tid] = 0.0f;

  // Ingest x with b128 loads/stores: 96 floats/row = 24 float4; rows are
  // 16B-aligned both in global (384B) and LDS (432B).
  {
    const float4* xb4 = (const float4*)(x + (size_t)blockIdx.x * (G * G));
    constexpr int Q_PER_ROW = G / 4;            // 24
    constexpr int NQ = G * Q_PER_ROW;           // 2304
    for (int q = tid; q < NQ; q += BLOCK) {
      const int r = q / Q_PER_ROW, c4 = q - r * Q_PER_ROW;
      *(float4*)(X + r * LDST + c4 * 4) = xb4[q];
    }
  }
  __syncthreads();

  wmma_gemm96(X, X, A2, wave, half, l);   // A2 = x^2
  __syncthreads();
  wmma_gemm96(X, A2, A3, wave, half, l);  // A3 = x^3
  __syncthreads();

  // t0..t4 = traces of x^2..x^6 as fused elementwise reductions
  float t0 = 0.f, t1 = 0.f, t2 = 0.f, t3 = 0.f, t4 = 0.f;
  for (int idx = tid; idx < G * G; idx += BLOCK) {
    const int r = idx / G, c = idx - r * G;
    const float xrc  = X[r * LDST + c],  xcr  = X[c * LDST + r];
    const float a2rc = A2[r * LDST + c], a2cr = A2[c * LDST + r];
    const float a3rc = A3[r * LDST + c], a3cr = A3[c * LDST + r];
    t0 += xrc * xcr;     // tr(x^2)
    t1 += xrc * a2cr;    // tr(x^3)
    t2 += a2rc * a2cr;   // tr(x^4)
    t3 += a2rc * a3cr;   // tr(x^5)
    t4 += a3rc * a3cr;   // tr(x^6)
  }
#pragma unroll
  for (int off = 16; off > 0; off >>= 1) {
    t0 += __shfl_down(t0, off, 32);
    t1 += __shfl_down(t1, off, 32);
    t2 += __shfl_down(t2, off, 32);
    t3 += __shfl_down(t3, off, 32);
    t4 += __shfl_down(t4, off, 32);
  }
  if (lane == 0) {
    atomicAdd(&T[0], t0);
    atomicAdd(&T[1], t1);
    atomicAdd(&T[2], t2);
    atomicAdd(&T[3], t3);
    atomicAdd(&T[4], t4);
  }
  __syncthreads();

  if (tid == 0) {
    const float numel = (float)(G * G);
    float dbase = numel;  // numel^(i+1) at j=0
    float res = 0.f;
    for (int i = 0; i < ROWS; ++i) {
      const float ti = T[i];
      float p = 1.f;       // ti^(j+1)
      float d = dbase;     // numel^(i+j+1)
      for (int j = 0; j < COLS; ++j) {
        p *= ti;
        res += coef[i * COLS + j] * p / d;
        d *= numel;
      }
      dbase *= numel;
    }
    out[blockIdx.x] = res;
  }
}

extern "C" void kernel_launch(void* const* d_in, const int* in_sizes, int n_in,
                              void* d_out, int out_size, void* d_ws, size_t ws_size,
                              hipStream_t stream) {
  const float* x    = (const float*)d_in[0];
  const float* coef = (const float*)d_in[1];
  float* out        = (float*)d_out;
  const int batch   = in_sizes[0] / (G * G);  // 2048
  const size_t shmem = (size_t)(3 * MATF + 8) * sizeof(float);  // ~124.5 KB (<= 320 KB WGP LDS)
  acoef_kernel<<<batch, BLOCK, shmem, stream>>>(x, coef, out);
}